// NCC2_loss_27642409517137
// MI455X (gfx1250) — compile-verified
//
#include <hip/hip_runtime.h>

typedef __attribute__((ext_vector_type(2))) float v2f;
typedef __attribute__((ext_vector_type(8))) float v8f;

#define S   160
#define S2  (160 * 160)
#define S3  (160 * 160 * 160)
#define NB  2
#define TOT (NB * S3)                 // 8,192,000 voxels
#define INV_VOL (1.0f / 729.0f)
#define EPS 1e-5f
#define HT  40                        // 32x32 block tile + 4-wide halo

// band(m, j) == 1 iff tile-local input row j (axis pos = p0-4+j) contributes
// to output row m (axis pos = p0+m):  m <= j <= m+8  (9-tap box).
__device__ __forceinline__ float band(int m, int j) {
    return ((unsigned)(j - m) <= 8u) ? 1.0f : 0.0f;
}

// ---------------------------------------------------------------------------
// Pass XY (fused): pointwise products + X box filter (from LDS) + Y box
// filter (banded-matrix WMMA).  One 128-thread block = 2x2 output tiles of
// 16y x 16x in one (n,z) plane; the block stages a zero-padded 40x40 I/J
// halo tile in LDS (branchless clamped loads).  Each wave then builds, per
// K-slice, the 5 channels' X-filtered B fragments straight from LDS (the 5
// window sums share one set of 18 LDS reads) and chains
//   D(16x16) = sum_{s=0..5} Wy_s(16x4) * B_s(4x16)     (fp32 WMMA 16x16x4)
// Fragment layouts per ISA 7.12.2: A lanes 0-15 hold K={0,1} in vgpr {0,1},
// lanes 16-31 hold K={2,3}; B vgpr0 = rows K0/K1 across lane halves, vgpr1 =
// rows K2/K3; C/D vgpr r = rows m=r / m=r+8, column n = lane&15.
// ---------------------------------------------------------------------------
__global__ __launch_bounds__(128) void pass_xy(const float* __restrict__ I,
                                               const float* __restrict__ J,
                                               float* __restrict__ B) {
    __shared__ float Ish[HT][HT];
    __shared__ float Jsh[HT][HT];

    const int tid  = threadIdx.x;
    const int lane = tid & 31;
    const int wave = tid >> 5;

    const int b   = blockIdx.x;            // 320 planes * 5 * 5 = 8000 blocks
    const int bx0 = (b % 5) * 32;
    const int by0 = ((b / 5) % 5) * 32;
    const int nz  = b / 25;                // combined (n,z) plane, 0..319
    const long plane = (long)nz * S2;

    // ---- stage 40x40 halo tile, zero-padded, branchless (clamp + select)
    for (int e = tid; e < HT * HT; e += 128) {
        const int r = e / HT;
        const int c = e % HT;
        const int y = by0 - 4 + r;
        const int x = bx0 - 4 + c;
        const bool ok = ((unsigned)y < (unsigned)S) && ((unsigned)x < (unsigned)S);
        const int yc = min(max(y, 0), S - 1);
        const int xc = min(max(x, 0), S - 1);
        const long a = plane + yc * S + xc;
        const float iv = I[a];
        const float jv = J[a];
        Ish[r][c] = ok ? iv : 0.0f;
        Jsh[r][c] = ok ? jv : 0.0f;
    }
    __syncthreads();

    // ---- per-wave 16x16 tile
    const int oy = (wave >> 1) * 16;       // tile origin inside block tile
    const int ox = (wave & 1) * 16;
    const int m  = lane & 15;              // A row (out y) / B-D column (x)
    const int hi = lane >> 4;

    v8f acc[5];
#pragma unroll
    for (int c = 0; c < 5; ++c) acc[c] = (v8f){0.f,0.f,0.f,0.f,0.f,0.f,0.f,0.f};

#pragma unroll
    for (int s = 0; s < 6; ++s) {
        const int kb = 4 * s;
        v2f a;
        a.x = band(m, kb + 2 * hi + 0);
        a.y = band(m, kb + 2 * hi + 1);

        v2f bf[5];
#pragma unroll
        for (int v = 0; v < 2; ++v) {
            const int j  = kb + 2 * v + hi;   // B K-row (y-halo row)
            const int ry = oy + j;            // row in staged tile
            const int cx = ox + m;            // leftmost tap column
            float s0 = 0.f, s1 = 0.f, s2 = 0.f, s3 = 0.f, s4 = 0.f;
#pragma unroll
            for (int k = 0; k < 9; ++k) {
                const float iv = Ish[ry][cx + k];
                const float jv = Jsh[ry][cx + k];
                s0 += iv;
                s1 += jv;
                s2 = fmaf(iv, iv, s2);
                s3 = fmaf(jv, jv, s3);
                s4 = fmaf(iv, jv, s4);
            }
            bf[0][v] = s0; bf[1][v] = s1; bf[2][v] = s2; bf[3][v] = s3; bf[4][v] = s4;
        }
#pragma unroll
        for (int c = 0; c < 5; ++c) {
            acc[c] = __builtin_amdgcn_wmma_f32_16x16x4_f32(
                false, a, false, bf[c], (short)0, acc[c], false, false);
        }
    }

    // ---- store XY-filtered fields (coalesced per half-wave)
#pragma unroll
    for (int c = 0; c < 5; ++c) {
        float* out = B + (long)c * TOT + plane;
#pragma unroll
        for (int r = 0; r < 8; ++r) {
            const int y = by0 + oy + r + 8 * hi;
            out[y * S + bx0 + ox + m] = acc[c][r];
        }
    }
}

// ---------------------------------------------------------------------------
// Pass Z: box filter along z (stride 160*160) for all 5 channels in one wave
// (banded WMMA, 5 accumulators), fused NCC combine + block reduction.
// Out-of-range z rows are handled branchlessly: the load address is clamped
// into [0,159] and the corresponding A-weight column is zeroed, so the
// clamped (garbage) values are annihilated and all loads are unconditional.
// ---------------------------------------------------------------------------
__global__ __launch_bounds__(256) void pass_z(const float* __restrict__ B,
                                              float* __restrict__ partial) {
    __shared__ float wsum[8];
    const int lane = threadIdx.x & 31;
    const int wave = threadIdx.x >> 5;
    const int t = blockIdx.x * 8 + wave;        // 32,000 tiles total
    const int tx = t % 10;
    const int tz = (t / 10) % 10;
    const int ny = t / 100;                     // 0..319
    const int n  = ny / S;
    const int y  = ny % S;
    const int m  = lane & 15;
    const int hi = lane >> 4;
    const int x  = tx * 16 + m;
    const int z0 = tz * 16;
    const long base = (long)n * S3 + (long)y * S;

    v8f acc[5];
#pragma unroll
    for (int c = 0; c < 5; ++c) acc[c] = (v8f){0.f,0.f,0.f,0.f,0.f,0.f,0.f,0.f};

#pragma unroll
    for (int s = 0; s < 6; ++s) {
        const int kb = 4 * s;
        // A weights: band * z-validity (zero column kills clamped loads)
        v2f a;
        {
            const int j0 = kb + 2 * hi + 0;
            const int j1 = kb + 2 * hi + 1;
            a.x = band(m, j0) * (((unsigned)(z0 - 4 + j0) < (unsigned)S) ? 1.0f : 0.0f);
            a.y = band(m, j1) * (((unsigned)(z0 - 4 + j1) < (unsigned)S) ? 1.0f : 0.0f);
        }
        // B rows: clamped, unconditional loads
        const int zA = min(max(z0 - 4 + (kb + 0 + hi), 0), S - 1);
        const int zB = min(max(z0 - 4 + (kb + 2 + hi), 0), S - 1);
#pragma unroll
        for (int c = 0; c < 5; ++c) {
            const float* in = B + (long)c * TOT + base;
            v2f bf;
            bf.x = in[(long)zA * S2 + x];
            bf.y = in[(long)zB * S2 + x];
            acc[c] = __builtin_amdgcn_wmma_f32_16x16x4_f32(
                false, a, false, bf, (short)0, acc[c], false, false);
        }
    }

    // Fused NCC combine on the accumulator fragments.
    float lsum = 0.0f;
#pragma unroll
    for (int r = 0; r < 8; ++r) {
        const float Is = acc[0][r], Js = acc[1][r];
        const float I2 = acc[2][r], J2 = acc[3][r], IJ = acc[4][r];
        const float cross = IJ - Is * Js * INV_VOL;
        const float Ivar  = I2 - Is * Is * INV_VOL;
        const float Jvar  = J2 - Js * Js * INV_VOL;
        lsum += cross * cross / (Ivar * Jvar + EPS);
    }
    // wave32 reduction
#pragma unroll
    for (int o = 16; o > 0; o >>= 1) lsum += __shfl_xor(lsum, o, 32);
    if (lane == 0) wsum[wave] = lsum;
    __syncthreads();
    if (threadIdx.x == 0) {
        float s = 0.0f;
#pragma unroll
        for (int w = 0; w < 8; ++w) s += wsum[w];
        partial[blockIdx.x] = s;
    }
}

__global__ __launch_bounds__(256) void final_reduce(const float* __restrict__ partial,
                                                    int n, float* __restrict__ out) {
    __shared__ double sh[256];
    double s = 0.0;
    for (int i = threadIdx.x; i < n; i += 256) s += (double)partial[i];
    sh[threadIdx.x] = s;
    __syncthreads();
    for (int o = 128; o > 0; o >>= 1) {
        if ((int)threadIdx.x < o) sh[threadIdx.x] += sh[threadIdx.x + o];
        __syncthreads();
    }
    if (threadIdx.x == 0) out[0] = (float)(-sh[0] / (double)TOT);
}

// ---------------------------------------------------------------------------
extern "C" void kernel_launch(void* const* d_in, const int* in_sizes, int n_in,
                              void* d_out, int out_size, void* d_ws, size_t ws_size,
                              hipStream_t stream) {
    const float* I = (const float*)d_in[0];
    const float* J = (const float*)d_in[1];
    float* ws = (float*)d_ws;
    float* Bf      = ws;                         // 5 * TOT floats (XY-filtered)
    float* partial = ws + (size_t)5 * TOT;       // 4000 floats

    pass_xy<<<NB * S * 25, 128, 0, stream>>>(I, J, Bf);          // 8,000 blocks
    pass_z<<<(NB * S * 100) / 8, 256, 0, stream>>>(Bf, partial); // 4,000 blocks
    final_reduce<<<1, 256, 0, stream>>>(partial, (NB * S * 100) / 8,
                                        (float*)d_out);
}